// QAdaptiveFeatureExtraction_46952582480158
// MI455X (gfx1250) — compile-verified
//
#include <hip/hip_runtime.h>
#include <math.h>

typedef float v2f __attribute__((ext_vector_type(2)));
typedef float v8f __attribute__((ext_vector_type(8)));

#define KC 32                   // input channels staged per LDS chunk
#define CH_LDS (3 * 66)         // 3 rows x 66 padded cols per channel
#define NPIX 4096               // 64*64
#define EPSV 1e-5f

// ---------------------------------------------------------------------------
// Stage A: global average pool over H,W.  One block per (b,c,q) -> 8192 blocks.
// x layout [B,C,Q,H,W]: block's 4096 elements are contiguous.
// ---------------------------------------------------------------------------
__global__ __launch_bounds__(256)
void k_pool_mean(const float* __restrict__ x, float* __restrict__ pooled) {
    __shared__ float s[256];
    long base = (long)blockIdx.x * NPIX;
    float sum = 0.f;
    for (int i = threadIdx.x; i < NPIX; i += 256) sum += x[base + i];
    s[threadIdx.x] = sum;
    __syncthreads();
    for (int o = 128; o > 0; o >>= 1) {
        if (threadIdx.x < (unsigned)o) s[threadIdx.x] += s[threadIdx.x + o];
        __syncthreads();
    }
    if (threadIdx.x == 0) pooled[blockIdx.x] = s[0] * (1.f / (float)NPIX);
}

// ---------------------------------------------------------------------------
// Stage B: global branch 1x1 conv + ReLU.  gfeat[b,o,q], 8*128*4 = 4096 outs.
// ---------------------------------------------------------------------------
__global__ __launch_bounds__(256)
void k_global_branch(const float* __restrict__ pooled, const float* __restrict__ wg,
                     const float* __restrict__ bg, float* __restrict__ gfeat,
                     int C, int Ch) {
    int idx = blockIdx.x * 256 + threadIdx.x;
    if (idx >= 8 * Ch * 4) return;
    int q = idx & 3;
    int o = (idx >> 2) % Ch;
    int b = idx / (Ch * 4);
    float s = bg[o];
    for (int c = 0; c < C; ++c)
        s = fmaf(wg[o * C + c], pooled[(b * C + c) * 4 + q], s);
    gfeat[idx] = fmaxf(s, 0.f);
}

// ---------------------------------------------------------------------------
// Stage C/F: 3x3 conv as implicit GEMM with V_WMMA_F32_16X16X4_F32.
// One workgroup (8 waves) per (image=b*4+q, output row).  Wave w owns output
// channels m0..m0+15 with m0 = (blockIdx.z*8 + w)*16; 4 accumulators cover the
// 64-pixel row.  K = C*9, staged KC channels at a time into LDS.
// ---------------------------------------------------------------------------
__device__ __forceinline__
void store_tile(const v8f& acc, int j, float* __restrict__ outb, long out_cs,
                int m0, int grp, int l2, int row, const float* __restrict__ bias) {
    int px = row * 64 + j * 16 + l2;
#pragma unroll
    for (int g = 0; g < 8; ++g) {
        int m = m0 + g + 8 * grp;              // C/D layout: VGPR g -> M=g / M=g+8
        outb[(long)m * out_cs + px] = acc[g] + bias[m];
    }
}

__global__ __launch_bounds__(256)
void k_conv3x3_wmma(const float* __restrict__ in, long in_bs, long in_qs, long in_cs,
                    const float* __restrict__ w, const float* __restrict__ bias,
                    float* __restrict__ out, long out_bs, long out_qs, long out_cs,
                    int C) {
    __shared__ float lds[KC * CH_LDS];          // 25,344 B

    const int img  = blockIdx.x;                // 0..31  (b*4+q)
    const int row  = blockIdx.y;                // 0..63  output row
    const int b    = img >> 2;
    const int q    = img & 3;
    const int tid  = threadIdx.x;
    const int wave = tid >> 5;
    const int lane = tid & 31;
    const int l2   = lane & 15;
    const int grp  = lane >> 4;                 // 0: K=0,1   1: K=2,3

    const float* inb  = in  + (long)b * in_bs  + (long)q * in_qs;
    float*       outb = out + (long)b * out_bs + (long)q * out_qs;

    const int m0 = (blockIdx.z * 8 + wave) * 16;
    const int K9 = C * 9;

    v8f acc0 = {}, acc1 = {}, acc2 = {}, acc3 = {};

    for (int c0 = 0; c0 < C; c0 += KC) {
        __syncthreads();
        // Stage KC channels x 3 rows x 66 cols (zero-padded halo) into LDS.
        for (int i = tid; i < KC * CH_LDS; i += 256) {
            int cl  = i / CH_LDS;
            int r   = i - cl * CH_LDS;
            int kh  = r / 66;
            int col = r - kh * 66;              // padded col: input col = col-1
            int irow = row + kh - 1;
            int icol = col - 1;
            float v = 0.f;
            if ((unsigned)irow < 64u && (unsigned)icol < 64u)
                v = inb[(long)(c0 + cl) * in_cs + irow * 64 + icol];
            lds[i] = v;
        }
        __syncthreads();

        const int ksteps = (KC * 9) / 4;        // 72 k-steps of 4
        for (int ks = 0; ks < ksteps; ++ks) {
            const int kbase = ks * 4 + 2 * grp; // this half-wave's first flat k
            // A fragment: weight rows m0+l2, flat K = c0*9 + kbase (+1)
            const float* wrow = w + (long)(m0 + l2) * K9 + (long)c0 * 9 + kbase;
            v2f a; a.x = wrow[0]; a.y = wrow[1];

            // B fragment: im2col from LDS; decompose the two flat-k values.
            int k0c = kbase / 9,  k0t = kbase - k0c * 9;
            int k1  = kbase + 1;
            int k1c = k1 / 9,     k1t = k1 - k1c * 9;
            int kh0 = k0t / 3,    kw0 = k0t - kh0 * 3;
            int kh1 = k1t / 3,    kw1 = k1t - kh1 * 3;
            int base0 = k0c * CH_LDS + kh0 * 66 + kw0;   // + pixel col
            int base1 = k1c * CH_LDS + kh1 * 66 + kw1;

            v2f b0, b1, b2, b3;
            b0.x = lds[base0 +  0 + l2]; b0.y = lds[base1 +  0 + l2];
            b1.x = lds[base0 + 16 + l2]; b1.y = lds[base1 + 16 + l2];
            b2.x = lds[base0 + 32 + l2]; b2.y = lds[base1 + 32 + l2];
            b3.x = lds[base0 + 48 + l2]; b3.y = lds[base1 + 48 + l2];

            acc0 = __builtin_amdgcn_wmma_f32_16x16x4_f32(false, a, false, b0, (short)0, acc0, false, false);
            acc1 = __builtin_amdgcn_wmma_f32_16x16x4_f32(false, a, false, b1, (short)0, acc1, false, false);
            acc2 = __builtin_amdgcn_wmma_f32_16x16x4_f32(false, a, false, b2, (short)0, acc2, false, false);
            acc3 = __builtin_amdgcn_wmma_f32_16x16x4_f32(false, a, false, b3, (short)0, acc3, false, false);
        }
    }

    store_tile(acc0, 0, outb, out_cs, m0, grp, l2, row, bias);
    store_tile(acc1, 1, outb, out_cs, m0, grp, l2, row, bias);
    store_tile(acc2, 2, outb, out_cs, m0, grp, l2, row, bias);
    store_tile(acc3, 3, outb, out_cs, m0, grp, l2, row, bias);
}

// ---------------------------------------------------------------------------
// IQBN statistics: one block per (co,q) pair; reduce over B=8 x 4096 pixels.
// Address of element: src + co*co_stride + q*q_stride + b*b_stride + pix.
// ---------------------------------------------------------------------------
__global__ __launch_bounds__(256)
void k_bn_stats(const float* __restrict__ src, long co_stride, long q_stride, long b_stride,
                float* __restrict__ mean, float* __restrict__ invstd) {
    __shared__ float ss[256], sq[256];
    int blk = blockIdx.x;                       // co*4 + q
    int q  = blk & 3;
    int co = blk >> 2;
    const float* p = src + (long)co * co_stride + (long)q * q_stride;
    float s = 0.f, s2 = 0.f;
    for (int b = 0; b < 8; ++b) {
        const float* pb = p + (long)b * b_stride;
        for (int i = threadIdx.x; i < NPIX; i += 256) {
            float v = pb[i];
            s += v; s2 = fmaf(v, v, s2);
        }
    }
    ss[threadIdx.x] = s; sq[threadIdx.x] = s2;
    __syncthreads();
    for (int o = 128; o > 0; o >>= 1) {
        if (threadIdx.x < (unsigned)o) {
            ss[threadIdx.x] += ss[threadIdx.x + o];
            sq[threadIdx.x] += sq[threadIdx.x + o];
        }
        __syncthreads();
    }
    if (threadIdx.x == 0) {
        const float inv_n = 1.f / 32768.f;      // 8 * 4096
        float m = ss[0] * inv_n;
        float v = sq[0] * inv_n - m * m;
        mean[blk]   = m;
        invstd[blk] = rsqrtf(v + EPSV);
    }
}

// ---------------------------------------------------------------------------
// Stage D: finalize combined [32,256,4096]:
//   c < 128 : ReLU(IQBN(local conv raw))           (in place)
//   c >= 128: broadcast gfeat[b, c-128, q]
// ---------------------------------------------------------------------------
__global__ __launch_bounds__(256)
void k_finalize_combined(float* __restrict__ comb, const float* __restrict__ gfeat,
                         const float* __restrict__ mean, const float* __restrict__ invstd,
                         const float* __restrict__ g, const float* __restrict__ be) {
    long idx = (long)blockIdx.x * 256 + threadIdx.x;
    if (idx >= 33554432L) return;               // 32*256*4096
    int c   = (int)((idx >> 12) & 255);
    int img = (int)(idx >> 20);
    int q = img & 3, b = img >> 2;
    if (c < 128) {
        int pr = c * 4 + q;
        float v = comb[idx];
        comb[idx] = fmaxf((v - mean[pr]) * invstd[pr] * g[c] + be[c], 0.f);
    } else {
        comb[idx] = gfeat[(b * 128 + (c - 128)) * 4 + q];
    }
}

// ---------------------------------------------------------------------------
// Stage E: channel attention, in place on combined.
// One block per 16 pixels: stage the 256x16 column tile in LDS, compute
// h = ReLU(W1 v + b1) (32x16), attn = sigmoid(W2 h + b2), v *= attn.
// ---------------------------------------------------------------------------
__global__ __launch_bounds__(256)
void k_channel_attention(float* __restrict__ comb,
                         const float* __restrict__ w1, const float* __restrict__ b1,
                         const float* __restrict__ w2, const float* __restrict__ b2) {
    __shared__ float v[256][16];
    __shared__ float h[32][16];
    int img  = blockIdx.x >> 8;                 // 0..31
    int tile = blockIdx.x & 255;                // 0..255
    int p0   = tile * 16;
    int t    = threadIdx.x;

    long base = ((long)img * 256 + t) * NPIX + p0;
#pragma unroll
    for (int j = 0; j < 16; ++j) v[t][j] = comb[base + j];
    __syncthreads();

    for (int oj = t; oj < 512; oj += 256) {     // 32 x 16 outputs of h
        int o = oj >> 4, j = oj & 15;
        float s = b1[o];
        for (int c = 0; c < 256; ++c) s = fmaf(w1[o * 256 + c], v[c][j], s);
        h[o][j] = fmaxf(s, 0.f);
    }
    __syncthreads();

#pragma unroll
    for (int j = 0; j < 16; ++j) {
        float s = b2[t];
#pragma unroll
        for (int k = 0; k < 32; ++k) s = fmaf(w2[t * 32 + k], h[k][j], s);
        float a = 1.f / (1.f + expf(-s));
        comb[base + j] = v[t][j] * a;
    }
}

// ---------------------------------------------------------------------------
// Stage G: final IQBN + ReLU, in place on d_out ([B,C,Q,H,W]).
// ---------------------------------------------------------------------------
__global__ __launch_bounds__(256)
void k_final_norm(float* __restrict__ out, const float* __restrict__ mean,
                  const float* __restrict__ invstd, const float* __restrict__ g,
                  const float* __restrict__ be) {
    long idx = (long)blockIdx.x * 256 + threadIdx.x;
    if (idx >= 33554432L) return;
    int q = (int)((idx >> 12) & 3);
    int c = (int)((idx >> 14) & 255);
    int pr = c * 4 + q;
    float v = out[idx];
    out[idx] = fmaxf((v - mean[pr]) * invstd[pr] * g[c] + be[c], 0.f);
}

// ---------------------------------------------------------------------------
extern "C" void kernel_launch(void* const* d_in, const int* in_sizes, int n_in,
                              void* d_out, int out_size, void* d_ws, size_t ws_size,
                              hipStream_t stream) {
    const float* x        = (const float*)d_in[0];
    const float* w_local  = (const float*)d_in[1];
    const float* b_local  = (const float*)d_in[2];
    const float* g_local  = (const float*)d_in[3];
    const float* be_local = (const float*)d_in[4];
    const float* w_global = (const float*)d_in[5];
    const float* b_global = (const float*)d_in[6];
    const float* w_ca1    = (const float*)d_in[7];
    const float* b_ca1    = (const float*)d_in[8];
    const float* w_ca2    = (const float*)d_in[9];
    const float* b_ca2    = (const float*)d_in[10];
    const float* w_refine = (const float*)d_in[11];
    const float* b_refine = (const float*)d_in[12];
    const float* g_norm   = (const float*)d_in[13];
    const float* be_norm  = (const float*)d_in[14];
    float* out = (float*)d_out;

    float* ws       = (float*)d_ws;
    float* comb     = ws;                       // 32*256*4096 = 33,554,432 floats
    float* pooled   = comb + 33554432L;         // 8192
    float* gfeat    = pooled + 8192;            // 4096
    float* mean_l   = gfeat + 4096;             // 512
    float* invstd_l = mean_l + 512;             // 512
    float* mean_f   = invstd_l + 512;           // 1024
    float* invstd_f = mean_f + 1024;            // 1024

    // strides (in floats)
    const long X_BS = 4194304, X_QS = 4096,    X_CS = 16384;   // x / d_out [B,C,Q,HW]
    const long C_BS = 4194304, C_QS = 1048576, C_CS = 4096;    // comb [img,C,HW]

    // A: pooled means
    k_pool_mean<<<8192, 256, 0, stream>>>(x, pooled);
    // B: global branch
    k_global_branch<<<16, 256, 0, stream>>>(pooled, w_global, b_global, gfeat, 256, 128);
    // C: local 3x3 conv (256 -> 128) into comb channels 0..127 (raw + bias)
    k_conv3x3_wmma<<<dim3(32, 64, 1), 256, 0, stream>>>(
        x, X_BS, X_QS, X_CS, w_local, b_local, comb, C_BS, C_QS, C_CS, 256);
    // local IQBN stats over (B,H,W) per (co<128, q)
    k_bn_stats<<<512, 256, 0, stream>>>(comb, C_CS, C_QS, C_BS, mean_l, invstd_l);
    // D: normalize local + ReLU, fill global-broadcast channels
    k_finalize_combined<<<131072, 256, 0, stream>>>(comb, gfeat, mean_l, invstd_l,
                                                    g_local, be_local);
    // E: channel attention in place
    k_channel_attention<<<8192, 256, 0, stream>>>(comb, w_ca1, b_ca1, w_ca2, b_ca2);
    // F: refine 3x3 conv (256 -> 256) into d_out (raw + bias)
    k_conv3x3_wmma<<<dim3(32, 64, 2), 256, 0, stream>>>(
        comb, C_BS, C_QS, C_CS, w_refine, b_refine, out, X_BS, X_QS, X_CS, 256);
    // final IQBN stats per (co<256, q)
    k_bn_stats<<<1024, 256, 0, stream>>>(out, X_CS, X_QS, X_BS, mean_f, invstd_f);
    // G: final normalize + ReLU in place
    k_final_norm<<<131072, 256, 0, stream>>>(out, mean_f, invstd_f, g_norm, be_norm);
}